// MLDRLoss_56160992362591
// MI455X (gfx1250) — compile-verified
//
#include <hip/hip_runtime.h>
#include <math.h>
#include <stdint.h>

typedef __attribute__((ext_vector_type(2))) float v2f;
typedef __attribute__((ext_vector_type(8))) float v8f;

#define T_LEN     524288
#define BATCH     16
#define ROWS      32                    // batch * (mid,side)
#define CHUNK     1024
#define NCHUNK    (T_LEN / CHUNK)       // 512
#define CPB       8                     // chunks per block in stage 3 (pipelined)
#define NFLAT     (ROWS * T_LEN)        // 16777216
#define EPS       1e-8f
#define INV_SQRT2 0.70710678118654752f

// ---------- small helpers ----------
// d^e for 0 <= e < 32, branchless binary exponentiation (no divergent loops)
__device__ __forceinline__ float fpowi(float d, int e) {
    float r = 1.f, b = d;
    #pragma unroll
    for (int i = 0; i < 5; ++i) {
        r = (e & 1) ? r * b : r;
        b *= b;
        e >>= 1;
    }
    return r;
}
__device__ __forceinline__ float pow2k(float d, int k) {  // d^(2^k)
    float r = d;
    for (int i = 0; i < k; ++i) r *= r;
    return r;
}
__device__ __forceinline__ float lane_read(float v, int srcLane) {
    return __int_as_float(__builtin_amdgcn_readlane(__float_as_int(v), srcLane));
}

// ---------- CDNA5 async global->LDS copy (ASYNCcnt path) ----------
__device__ __forceinline__ void async_copy_b128(const float* gsrc, float* ldst) {
    uint64_t ga = (uint64_t)(uintptr_t)gsrc;
    uint32_t la = (uint32_t)(uintptr_t)ldst;   // low 32 bits of generic LDS ptr = LDS addr
    asm volatile("global_load_async_to_lds_b128 %0, %1, off"
                 :: "v"(la), "v"(ga) : "memory");
}
template<int N>
__device__ __forceinline__ void wait_async_le() {
    asm volatile("s_wait_asynccnt %0" :: "i"(N) : "memory");
}

// Issue the 4-stream staging of one (batch, chunk): 4 async b128 per thread.
__device__ __forceinline__ void issue_chunk_loads(const float* __restrict__ xp,
                                                  const float* __restrict__ xt,
                                                  int b, int chunk,
                                                  float (*raw)[CHUNK]) {
    const int tid = threadIdx.x;                      // 256 threads
    const size_t base0 = (size_t)(b * 2 + 0) * T_LEN + (size_t)chunk * CHUNK;
    const size_t base1 = (size_t)(b * 2 + 1) * T_LEN + (size_t)chunk * CHUNK;
    const float* srcs[4] = { xp + base0, xp + base1, xt + base0, xt + base1 };
    #pragma unroll
    for (int s = 0; s < 4; ++s)
        async_copy_b128(srcs[s] + tid * 4, &raw[s][tid * 4]);
}

// Mid/side squares, clamped: sq[0]=pred-mid sq[1]=pred-side sq[2]=true-mid sq[3]=true-side
__device__ __forceinline__ void compute_squares(const float (*raw)[CHUNK],
                                                float (*sq)[CHUNK]) {
    const int j = threadIdx.x * 4;   // 1024 floats / 256 threads = 4 each
    #pragma unroll
    for (int u = 0; u < 4; ++u) {
        float pl = raw[0][j + u], pr = raw[1][j + u];
        float tl = raw[2][j + u], tr = raw[3][j + u];
        float pm = (pl + pr) * INV_SQRT2, ps = (pl - pr) * INV_SQRT2;
        float tm = (tl + tr) * INV_SQRT2, ts = (tl - tr) * INV_SQRT2;
        sq[0][j + u] = fmaxf(pm * pm, EPS);
        sq[1][j + u] = fmaxf(ps * ps, EPS);
        sq[2][j + u] = fmaxf(tm * tm, EPS);
        sq[3][j + u] = fmaxf(ts * ts, EPS);
    }
}

// ---------- Stage 1: per-chunk EMA tails (zero-carry local scan end value) ----
__global__ __launch_bounds__(256)
void k_chunk_tails(const float* __restrict__ xp, const float* __restrict__ xt,
                   float* __restrict__ tails, float cS, float cL) {
    __shared__ float raw[4][CHUNK];
    __shared__ float sq[4][CHUNK];
    const int b = blockIdx.x / NCHUNK;
    const int chunk = blockIdx.x % NCHUNK;
    issue_chunk_loads(xp, xt, b, chunk, raw);
    wait_async_le<0>();
    __syncthreads();
    compute_squares(raw, sq);
    __syncthreads();

    const int w    = threadIdx.x >> 5;   // 8 waves <-> 8 scan lines
    const int lane = threadIdx.x & 31;
    const int m  = (w >> 2) & 1;         // mid/side
    const int sg = (w >> 1) & 1;         // 0=pred 1=true
    const int co = w & 1;                // 0=short 1=long
    const float c = co ? cL : cS;
    const float d = 1.f - c;
    const float* x = sq[sg * 2 + m];

    // lane covers 32 consecutive samples: weighted suffix sum
    float s = 0.f;
    const int sb = lane * 32;
    for (int i = 0; i < 32; ++i) s = s * d + x[sb + i];
    const float D32 = pow2k(d, 5);            // d^32
    float v = s * fpowi(D32, 31 - lane);      // branchless per-lane weight
    for (int off = 16; off > 0; off >>= 1) v += __shfl_xor(v, off, 32);
    if (lane == 0) {
        const int line = (b * 2 + m) * 4 + sg * 2 + co;
        tails[(size_t)line * NCHUNK + chunk] = c * v;
    }
}

// ---------- Stage 2: inter-chunk carry scan (128 lines x 512 chunks) ---------
__global__ __launch_bounds__(128)
void k_scan_carries(const float* __restrict__ tails, float* __restrict__ carries,
                    float cS, float cL) {
    const int line = threadIdx.x;        // 0..127
    const int co = line & 1;
    const float d = 1.f - (co ? cL : cS);
    const float Dc = pow2k(d, 10);       // d^CHUNK = d^1024
    float carry = 0.f;
    const size_t off = (size_t)line * NCHUNK;
    for (int k = 0; k < NCHUNK; ++k) {
        carries[off + k] = carry;
        carry = tails[off + k] + Dc * carry;
    }
}

// ---------- Stage 3: pipelined WMMA scan + log + field combine ---------------
__global__ __launch_bounds__(256)
void k_ema_log_fields(const float* __restrict__ xp, const float* __restrict__ xt,
                      const float* __restrict__ carries,
                      float* __restrict__ Af, float* __restrict__ Bf,
                      float cS, float cL) {
    __shared__ float raw[2][4][CHUNK];   // double-buffered async staging
    __shared__ float sq[4][CHUNK];
    __shared__ float logb[8][CHUNK];
    const int b  = blockIdx.x / (NCHUNK / CPB);
    const int c0 = (blockIdx.x % (NCHUNK / CPB)) * CPB;

    const int w    = threadIdx.x >> 5;
    const int lane = threadIdx.x & 31;
    const int m  = (w >> 2) & 1;
    const int sg = (w >> 1) & 1;
    const int co = w & 1;
    const float c = co ? cL : cS;
    const float d = 1.f - c;
    const float* x = sq[sg * 2 + m];
    const size_t lineOff = (size_t)((b * 2 + m) * 4 + sg * 2 + co) * NCHUNK;

    // A operand: constant 16x16 lower-triangular scan matrix L[t][j]=c*d^(t-j),
    // split into 4 K-slices of the 16x4 f32 A layout. Built once per block.
    const int M = lane & 15;
    const int o = (lane >= 16) ? 2 : 0;
    v2f A[4];
    #pragma unroll
    for (int k = 0; k < 4; ++k) {
        const int j0 = 4 * k + o, j1 = j0 + 1;
        A[k].x = (j0 <= M) ? c * fpowi(d, M - j0) : 0.f;
        A[k].y = (j1 <= M) ? c * fpowi(d, M - j1) : 0.f;
    }
    const float D16 = fpowi(d, 16);
    const int n = lane & 15;                 // output column (16-sample block id)
    const int rowoff = (lane >= 16) ? 8 : 0; // C/D rows 0..7 vs 8..15
    const float dp0 = fpowi(d, rowoff + 1);
    const int jv = threadIdx.x * 4;          // combine-store lane offset

    issue_chunk_loads(xp, xt, b, c0, raw[0]);      // prime the pipeline

    for (int cc = 0; cc < CPB; ++cc) {
        const int chunk = c0 + cc;
        if (cc + 1 < CPB) {
            issue_chunk_loads(xp, xt, b, chunk + 1, raw[(cc + 1) & 1]);
            wait_async_le<4>();    // only the OLDER group of 4 must be done
        } else {
            wait_async_le<0>();
        }
        __syncthreads();
        compute_squares(raw[cc & 1], sq);
        __syncthreads();

        float carry = carries[lineOff + chunk];

        #pragma unroll
        for (int tile = 0; tile < 4; ++tile) {   // 4 x 256 samples per chunk
            const int tb = tile * 256;
            v8f acc = {};
            #pragma unroll
            for (int k = 0; k < 4; ++k) {
                // B operand (4x16 K-slice): X[kk][n] = x[n*16+kk]
                const int a = tb + n * 16 + 4 * k + o;   // even -> 8B aligned
                const v2f Bv = *(const v2f*)(&x[a]);
                acc = __builtin_amdgcn_wmma_f32_16x16x4_f32(
                          false, A[k], false, Bv, (short)0, acc, false, false);
            }
            // propagate carries across 16 columns: S_{mc+1} = yend_mc + d^16 * S_mc
            // yend[mc] = Y[15][mc], held in lane 16+mc, acc[7]  ->  v_readlane
            float S = carry, full = carry;
            #pragma unroll
            for (int mc = 0; mc < 16; ++mc) {
                const float ye = lane_read(acc[7], 16 + mc);
                S = (n > mc) ? ye + D16 * S : S;
                full = ye + D16 * full;
            }
            // finalize: y[t][n] = Y[t][n] + d^(t+1) * S_n ; take log
            float dp = dp0;
            #pragma unroll
            for (int r = 0; r < 8; ++r) {
                const float y = acc[r] + dp * S;
                logb[w][tb + n * 16 + rowoff + r] = __logf(y);
                dp *= d;
            }
            carry = full;
        }
        __syncthreads();

        // Combine: A = logS(pred)-logS(true), B = logL(pred)-logL(true)
        #pragma unroll
        for (int mm = 0; mm < 2; ++mm) {
            const size_t gbase = (size_t)(b * 2 + mm) * T_LEN + (size_t)chunk * CHUNK;
            const float4 spv = *(const float4*)&logb[mm * 4 + 0][jv];
            const float4 lpv = *(const float4*)&logb[mm * 4 + 1][jv];
            const float4 stv = *(const float4*)&logb[mm * 4 + 2][jv];
            const float4 ltv = *(const float4*)&logb[mm * 4 + 3][jv];
            float4 av, bv;
            av.x = spv.x - stv.x; av.y = spv.y - stv.y;
            av.z = spv.z - stv.z; av.w = spv.w - stv.w;
            bv.x = lpv.x - ltv.x; bv.y = lpv.y - ltv.y;
            bv.z = lpv.z - ltv.z; bv.w = lpv.w - ltv.w;
            *(float4*)&Af[gbase + jv] = av;
            *(float4*)&Bf[gbase + jv] = bv;
        }
        __syncthreads();
    }
}

// ---------- Stage 4: |A[i] - B[(i+shift) mod N]| mean-reduce ------------------
__global__ __launch_bounds__(256)
void k_reduce_abs(const float* __restrict__ Af, const float* __restrict__ Bf,
                  int shift, float* __restrict__ out) {
    const float4* Af4 = (const float4*)Af;
    float acc = 0.f;
    const int stride = gridDim.x * blockDim.x;
    for (int i4 = blockIdx.x * blockDim.x + threadIdx.x; i4 < NFLAT / 4; i4 += stride) {
        const float4 a = Af4[i4];
        const int jb = i4 * 4 + shift;                 // shift is odd -> scalar B side
        int j0 = jb + 0; j0 = (j0 >= NFLAT) ? j0 - NFLAT : j0;
        int j1 = jb + 1; j1 = (j1 >= NFLAT) ? j1 - NFLAT : j1;
        int j2 = jb + 2; j2 = (j2 >= NFLAT) ? j2 - NFLAT : j2;
        int j3 = jb + 3; j3 = (j3 >= NFLAT) ? j3 - NFLAT : j3;
        acc += fabsf(a.x - Bf[j0]) + fabsf(a.y - Bf[j1]) +
               fabsf(a.z - Bf[j2]) + fabsf(a.w - Bf[j3]);
    }
    for (int off = 16; off > 0; off >>= 1) acc += __shfl_xor(acc, off, 32);
    __shared__ float wsum[8];
    const int lane = threadIdx.x & 31, w = threadIdx.x >> 5;
    if (lane == 0) wsum[w] = acc;
    __syncthreads();
    if (threadIdx.x == 0) {
        float s = 0.f;
        for (int i = 0; i < 8; ++i) s += wsum[i];
        atomicAdd(out, s * (1.0f / (float)NFLAT));
    }
}

__global__ void k_zero(float* out) { out[0] = 0.f; }

// ---------- host ----------
extern "C" void kernel_launch(void* const* d_in, const int* in_sizes, int n_in,
                              void* d_out, int out_size, void* d_ws, size_t ws_size,
                              hipStream_t stream) {
    (void)in_sizes; (void)n_in; (void)out_size; (void)ws_size;
    const float* xp = (const float*)d_in[0];
    const float* xt = (const float*)d_in[1];
    float* out = (float*)d_out;

    char* ws = (char*)d_ws;
    float* Af      = (float*)ws;                                  // N floats
    float* Bf      = (float*)(ws + (size_t)NFLAT * 4);            // N floats
    float* tails   = (float*)(ws + (size_t)NFLAT * 8);            // 128*512
    float* carries = tails + (size_t)ROWS * 4 * NCHUNK;           // 128*512

    // coef = 1 - exp(-2200/(ms*44100)); shift = int(44100*(l-s)*0.0005)
    static const float CS[3] = {4.9762393e-3f, 9.9723478e-4f, 4.9874178e-4f};
    static const float CL[3] = {4.9885377e-5f, 1.6628735e-5f, 9.9772745e-6f};
    static const int   SH[3] = {21829, 65047, 108045};

    k_zero<<<dim3(1), dim3(1), 0, stream>>>(out);
    for (int s = 0; s < 3; ++s) {
        k_chunk_tails<<<dim3(BATCH * NCHUNK), dim3(256), 0, stream>>>(
            xp, xt, tails, CS[s], CL[s]);
        k_scan_carries<<<dim3(1), dim3(128), 0, stream>>>(
            tails, carries, CS[s], CL[s]);
        k_ema_log_fields<<<dim3(BATCH * (NCHUNK / CPB)), dim3(256), 0, stream>>>(
            xp, xt, carries, Af, Bf, CS[s], CL[s]);
        k_reduce_abs<<<dim3(2048), dim3(256), 0, stream>>>(Af, Bf, SH[s], out);
    }
}